// LayerGINConv_7430293422228
// MI455X (gfx1250) — compile-verified
//
#include <hip/hip_runtime.h>

// ---------------------------------------------------------------------------
// GINConv(50000 nodes, 800000 edges, 128ch) -> Linear -> BN(train) -> ReLU ->
// Linear, fp32 end-to-end using V_WMMA_F32_16X16X4_F32 on gfx1250 (wave32).
// ---------------------------------------------------------------------------

#define NN 50000
#define NE 800000
#define DCH 128
#define BN_EPS 1e-5f

typedef __attribute__((ext_vector_type(2))) float v2f;
typedef __attribute__((ext_vector_type(8))) float v8f;

// ---------------------------------------------------------------------------
// Kernel 1: h0 = x  (EPS_GIN == 0 so combine init is just a copy),
// and zero the 256-float stats accumulators (colsum[128], colsumsq[128]).
// ---------------------------------------------------------------------------
__global__ __launch_bounds__(256) void gin_init_kernel(
    const float* __restrict__ x, float* __restrict__ h0,
    float* __restrict__ stats) {
  int tid = blockIdx.x * 256 + threadIdx.x;
  if (blockIdx.x == 0) stats[threadIdx.x] = 0.0f;  // 256 accumulators
  const float4* xv = (const float4*)x;
  float4* hv = (float4*)h0;
  if (tid < NN * DCH / 4) hv[tid] = xv[tid];
}

// ---------------------------------------------------------------------------
// Kernel 2: scatter-add  h0[dst] += x[src].  32 lanes per edge, each lane
// owns 4 consecutive channels: one b128 load + 4 global_atomic_add_f32.
// ---------------------------------------------------------------------------
__global__ __launch_bounds__(256) void gin_scatter_kernel(
    const float* __restrict__ x, const int* __restrict__ src,
    const int* __restrict__ dst, float* __restrict__ h0) {
  unsigned tid = blockIdx.x * 256u + threadIdx.x;
  unsigned e = tid >> 5;
  if (e >= NE) return;
  int c4 = (tid & 31) * 4;
  int s = src[e];
  int d = dst[e];
  float4 v = *(const float4*)(x + (size_t)s * DCH + c4);
  float* p = h0 + (size_t)d * DCH + c4;
  atomicAdd(p + 0, v.x);
  atomicAdd(p + 1, v.y);
  atomicAdd(p + 2, v.z);
  atomicAdd(p + 3, v.w);
}

// ---------------------------------------------------------------------------
// GEMM1: h1 = h0 @ W1 + b1, fused column sum / sum-of-squares for BN stats.
// One wave computes one 16x16 output tile; block = 8 waves sharing one
// 16-row A stripe (tileM = blockIdx.x, tileN = wave).  K loop: 32 x
// V_WMMA_F32_16X16X4_F32.
//
// A fragment (16x4 fp32, ISA 7.12.2): lane L<16  -> (M=L,    K={k,k+1})
//                                     lane L>=16 -> (M=L-16, K={k+2,k+3})
// B fragment (4x16 fp32, mirrored):   lane L holds (K=k+2*(L>>4)+v, N=L&15)
// C/D (16x16 fp32): vgpr i -> row (L<16 ? i : 8+i), col = L&15.
// ---------------------------------------------------------------------------
__global__ __launch_bounds__(256) void gin_gemm1_kernel(
    const float* __restrict__ h0, const float* __restrict__ W1,
    const float* __restrict__ b1, float* __restrict__ h1,
    float* __restrict__ colsum, float* __restrict__ colsumsq) {
  const int lane = threadIdx.x & 31;
  const int wave = threadIdx.x >> 5;
  const int tileM = blockIdx.x;  // 0..3124 (50000/16 exact)
  const int tileN = wave;        // 0..7    (128/16)
  const int r = lane & 15;
  const int halfK = lane >> 4;

  const float* aBase = h0 + (size_t)(tileM * 16 + r) * DCH + halfK * 2;
  const float* bBase = W1 + (size_t)(halfK * 2) * DCH + tileN * 16 + r;

  v8f acc = {0.f, 0.f, 0.f, 0.f, 0.f, 0.f, 0.f, 0.f};
#pragma unroll
  for (int k = 0; k < DCH; k += 4) {
    v2f a;
    a.x = aBase[k];
    a.y = aBase[k + 1];
    v2f b;
    b.x = bBase[(size_t)k * DCH];
    b.y = bBase[(size_t)(k + 1) * DCH];
    acc = __builtin_amdgcn_wmma_f32_16x16x4_f32(false, a, false, b, (short)0,
                                                acc, false, false);
  }

  const int col = tileN * 16 + r;
  const float bias = b1[col];
  const int rowBase = tileM * 16 + halfK * 8;
  float s = 0.f, sq = 0.f;
#pragma unroll
  for (int i = 0; i < 8; ++i) {
    float v = acc[i] + bias;
    h1[(size_t)(rowBase + i) * DCH + col] = v;
    s += v;
    sq += v * v;
  }
  atomicAdd(&colsum[col], s);
  atomicAdd(&colsumsq[col], sq);
}

// ---------------------------------------------------------------------------
// Kernel 4: fold batch stats into per-channel affine:
//   scale = gamma * rsqrt(var + eps),  shift = beta - mean*scale
// ---------------------------------------------------------------------------
__global__ __launch_bounds__(128) void gin_bnfold_kernel(
    const float* __restrict__ colsum, const float* __restrict__ colsumsq,
    const float* __restrict__ gamma, const float* __restrict__ beta,
    float* __restrict__ scale, float* __restrict__ shift) {
  int c = threadIdx.x;
  const float invN = 1.0f / (float)NN;
  float m = colsum[c] * invN;
  float var = colsumsq[c] * invN - m * m;
  float sc = rsqrtf(var + BN_EPS) * gamma[c];
  scale[c] = sc;
  shift[c] = beta[c] - m * sc;
}

// ---------------------------------------------------------------------------
// GEMM2: out = relu(h1*scale + shift) @ W2 + b2.  BN affine + ReLU fused
// into the A-fragment load; scale/shift staged through LDS.
// ---------------------------------------------------------------------------
__global__ __launch_bounds__(256) void gin_gemm2_kernel(
    const float* __restrict__ h1, const float* __restrict__ W2,
    const float* __restrict__ b2, const float* __restrict__ scale,
    const float* __restrict__ shift, float* __restrict__ out) {
  __shared__ float sScale[DCH];
  __shared__ float sShift[DCH];
  if (threadIdx.x < DCH) {
    sScale[threadIdx.x] = scale[threadIdx.x];
    sShift[threadIdx.x] = shift[threadIdx.x];
  }
  __syncthreads();

  const int lane = threadIdx.x & 31;
  const int wave = threadIdx.x >> 5;
  const int tileM = blockIdx.x;
  const int tileN = wave;
  const int r = lane & 15;
  const int halfK = lane >> 4;

  const float* aBase = h1 + (size_t)(tileM * 16 + r) * DCH + halfK * 2;
  const float* bBase = W2 + (size_t)(halfK * 2) * DCH + tileN * 16 + r;

  v8f acc = {0.f, 0.f, 0.f, 0.f, 0.f, 0.f, 0.f, 0.f};
#pragma unroll
  for (int k = 0; k < DCH; k += 4) {
    const int c0 = k + halfK * 2;
    v2f a;
    a.x = fmaxf(fmaf(aBase[k],     sScale[c0],     sShift[c0]),     0.f);
    a.y = fmaxf(fmaf(aBase[k + 1], sScale[c0 + 1], sShift[c0 + 1]), 0.f);
    v2f b;
    b.x = bBase[(size_t)k * DCH];
    b.y = bBase[(size_t)(k + 1) * DCH];
    acc = __builtin_amdgcn_wmma_f32_16x16x4_f32(false, a, false, b, (short)0,
                                                acc, false, false);
  }

  const int col = tileN * 16 + r;
  const float bias = b2[col];
  const int rowBase = tileM * 16 + halfK * 8;
#pragma unroll
  for (int i = 0; i < 8; ++i) {
    out[(size_t)(rowBase + i) * DCH + col] = acc[i] + bias;
  }
}

// ---------------------------------------------------------------------------
// Launch.  d_ws layout: h0[6.4M f32] | h1[6.4M f32] | colsum[128] |
// colsumsq[128] | scale[128] | shift[128]  (~51.2 MB).
// ---------------------------------------------------------------------------
extern "C" void kernel_launch(void* const* d_in, const int* in_sizes, int n_in,
                              void* d_out, int out_size, void* d_ws,
                              size_t ws_size, hipStream_t stream) {
  const float* x     = (const float*)d_in[0];
  const int*   ei    = (const int*)d_in[1];
  const float* W1    = (const float*)d_in[2];
  const float* b1    = (const float*)d_in[3];
  const float* gamma = (const float*)d_in[4];
  const float* beta  = (const float*)d_in[5];
  const float* W2    = (const float*)d_in[6];
  const float* b2    = (const float*)d_in[7];
  float* out = (float*)d_out;

  float* h0 = (float*)d_ws;
  float* h1 = h0 + (size_t)NN * DCH;
  float* stats = h1 + (size_t)NN * DCH;
  float* colsum = stats;
  float* colsumsq = stats + DCH;
  float* scale = stats + 2 * DCH;
  float* shift = stats + 3 * DCH;

  const int* src = ei;        // edge_index[0]
  const int* dst = ei + NE;   // edge_index[1]

  gin_init_kernel<<<(NN * DCH / 4 + 255) / 256, 256, 0, stream>>>(x, h0, stats);
  gin_scatter_kernel<<<(NE * 32 + 255) / 256, 256, 0, stream>>>(x, src, dst, h0);
  gin_gemm1_kernel<<<NN / 16, 256, 0, stream>>>(h0, W1, b1, h1, colsum, colsumsq);
  gin_bnfold_kernel<<<1, DCH, 0, stream>>>(colsum, colsumsq, gamma, beta, scale, shift);
  gin_gemm2_kernel<<<NN / 16, 256, 0, stream>>>(h1, W2, b2, scale, shift, out);
}